// ARMA_27212912788334
// MI455X (gfx1250) — compile-verified
//
#include <hip/hip_runtime.h>
#include <hip/hip_bf16.h>

// ---------------------------------------------------------------------------
// ARMA GNN layer for MI455X (gfx1250, wave32).
// fp32 throughout; matrix work on V_WMMA_F32_16X16X4_F32 so the ~10 GFLOP of
// GEMM stays far below the ~12-15us HBM roofline (250-350MB @ 23.3 TB/s).
// All 64-wide intermediates (~84MB) are L2-resident (192MB L2), so the
// gather/scatter propagation is L2 traffic, not HBM.
// ---------------------------------------------------------------------------

typedef float v2f __attribute__((ext_vector_type(2)));
typedef float v8f __attribute__((ext_vector_type(8)));

#define N_NODES 100000
#define N_EDGES 1600000
#define F_HID   64

__device__ __forceinline__ v8f wmma_f32x4(v2f a, v2f b, v8f c) {
  // 8-arg form: (neg_a, A, neg_b, B, c_mod, C, reuse_a, reuse_b)
  return __builtin_amdgcn_wmma_f32_16x16x4_f32(false, a, false, b, (short)0, c,
                                               false, false);
}

// --------------------------- tiny elementwise kernels ----------------------

__global__ void __launch_bounds__(256) zero_kernel(float* __restrict__ p, int n) {
  int t = blockIdx.x * 256 + threadIdx.x;
  if (t < n) p[t] = 0.0f;
}

__global__ void __launch_bounds__(256) dinv_kernel(float* __restrict__ deg, int n) {
  int t = blockIdx.x * 256 + threadIdx.x;
  if (t < n) {
    float d = deg[t];
    deg[t] = (d > 0.0f) ? rsqrtf(d) : 0.0f;  // deg buffer becomes dinv
  }
}

__global__ void __launch_bounds__(256)
norm_kernel(float* __restrict__ ew, const float* __restrict__ dinv,
            const int* __restrict__ row, const int* __restrict__ col) {
  int t = blockIdx.x * 256 + threadIdx.x;
  if (t < N_EDGES) {
    float w = ew[t];
    ew[t] = dinv[row[t]] * w * dinv[col[t]];  // ew buffer becomes norm
  }
}

// --------------------------- edge MLP (WMMA) -------------------------------
// One 16-edge tile per wave. Layer 1: [16x16]@[16x64] via 4 n-tiles x 4
// k-steps of WMMA f32 16x16x4 (W1 staged in LDS, rows padded to 68 floats).
// Layer 2 ([16x64]@[64x1]) via per-lane scale + 16-lane shfl_xor reduction.
// Fuses deg[col] += ew (segment_sum for gcn_norm).

__global__ void __launch_bounds__(256)
edge_mlp_kernel(const float* __restrict__ attr, const float* __restrict__ w1,
                const float* __restrict__ b1v, const float* __restrict__ w2,
                const float* __restrict__ b2v, const int* __restrict__ col,
                float* __restrict__ ew, float* __restrict__ deg) {
  __shared__ float sW1[16 * 68];
  __shared__ float sW2[64];
  const int tid = threadIdx.x;
  for (int i = tid; i < 16 * 64; i += 256)
    sW1[(i >> 6) * 68 + (i & 63)] = w1[i];
  if (tid < 64) sW2[tid] = w2[tid];
  __syncthreads();

  const int tile = (blockIdx.x * 256 + tid) >> 5;  // 8 waves / block
  if (tile >= N_EDGES / 16) return;
  const int lane = tid & 31;
  const int half = lane >> 4;
  const int l16  = lane & 15;
  const long base = (long)tile * 16;

  v8f c0 = {}, c1 = {}, c2 = {}, c3 = {};
  // A frag (16x4 fp32): lanes 0-15 hold K=ks+0/+1, lanes 16-31 K=ks+2/+3.
  const float* arow = attr + (base + l16) * 16 + half * 2;
#pragma unroll
  for (int ks = 0; ks < 16; ks += 4) {
    v2f A = {arow[ks], arow[ks + 1]};
    const float* wr = &sW1[(ks + half * 2) * 68 + l16];  // row k0 / k0+2
    v2f B0 = {wr[0],  wr[68]};       // cols 0..15, rows k0 / k0+1
    v2f B1 = {wr[16], wr[16 + 68]};
    v2f B2 = {wr[32], wr[32 + 68]};
    v2f B3 = {wr[48], wr[48 + 68]};
    c0 = wmma_f32x4(A, B0, c0);
    c1 = wmma_f32x4(A, B1, c1);
    c2 = wmma_f32x4(A, B2, c2);
    c3 = wmma_f32x4(A, B3, c3);
  }

  // layer 2: s[v] = sum_n relu(h[m][n] + b1[n]) * w2[n]
  float s[8];
#pragma unroll
  for (int v = 0; v < 8; ++v) s[v] = 0.0f;
#define ACC_TILE(CT, NT)                                             \
  {                                                                  \
    float bia = b1v[(NT)*16 + l16];                                  \
    float sc  = sW2[(NT)*16 + l16];                                  \
    _Pragma("unroll") for (int v = 0; v < 8; ++v)                    \
        s[v] += fmaxf(CT[v] + bia, 0.0f) * sc;                       \
  }
  ACC_TILE(c0, 0) ACC_TILE(c1, 1) ACC_TILE(c2, 2) ACC_TILE(c3, 3)
#undef ACC_TILE

  // reduce each row-sum across the 16-lane group (C rows v / v+8 per half)
#pragma unroll
  for (int v = 0; v < 8; ++v) {
    float t = s[v];
    t += __shfl_xor(t, 8, 16);
    t += __shfl_xor(t, 4, 16);
    t += __shfl_xor(t, 2, 16);
    t += __shfl_xor(t, 1, 16);
    s[v] = t;
  }
  float my = s[0];
#pragma unroll
  for (int v = 1; v < 8; ++v) my = (l16 == v) ? s[v] : my;
  if (l16 < 8) {
    float ewv = fmaxf(my + b2v[0], 0.0f);
    long e = base + half * 8 + l16;
    ew[e] = ewv;
    atomicAdd(&deg[col[e]], ewv);  // deg = segment_sum(ew, col)
  }
}

// --------------------------- node GEMM (WMMA) ------------------------------
// One 16-row tile per wave: out[16x64] = (relu?)(in[16xK]) @ w[Kx64] (+bias).
// A/B fragments per ISA 7.12.2 fp32 layouts; C/D: VGPR v = rows v / v+8.

template <int RELU_IN>
__global__ void __launch_bounds__(256)
gemm_node_kernel(const float* __restrict__ in, const float* __restrict__ w,
                 const float* __restrict__ bias, float* __restrict__ out,
                 int M, int K) {
  const int tid  = threadIdx.x;
  const int wave = (blockIdx.x * 256 + tid) >> 5;
  const int tiles = (M + 15) >> 4;
  if (wave >= tiles) return;
  const int lane = tid & 31;
  const int half = lane >> 4;
  const int l16  = lane & 15;
  const int row_base = wave << 4;
  int ra = row_base + l16;
  if (ra > M - 1) ra = M - 1;  // clamp loads; stores guarded below
  const float* arow = in + (long)ra * K + half * 2;

  v8f c0 = {}, c1 = {}, c2 = {}, c3 = {};
  for (int ks = 0; ks < K; ks += 4) {
    float a0 = arow[ks], a1 = arow[ks + 1];
    if (RELU_IN) { a0 = fmaxf(a0, 0.0f); a1 = fmaxf(a1, 0.0f); }
    v2f A = {a0, a1};
    const float* wr = w + (long)(ks + half * 2) * 64 + l16;
    v2f B0 = {wr[0],  wr[64]};
    v2f B1 = {wr[16], wr[80]};
    v2f B2 = {wr[32], wr[96]};
    v2f B3 = {wr[48], wr[112]};
    c0 = wmma_f32x4(A, B0, c0);
    c1 = wmma_f32x4(A, B1, c1);
    c2 = wmma_f32x4(A, B2, c2);
    c3 = wmma_f32x4(A, B3, c3);
  }

  const float bv0 = bias ? bias[l16]      : 0.0f;
  const float bv1 = bias ? bias[16 + l16] : 0.0f;
  const float bv2 = bias ? bias[32 + l16] : 0.0f;
  const float bv3 = bias ? bias[48 + l16] : 0.0f;
#pragma unroll
  for (int v = 0; v < 8; ++v) {
    int r = row_base + half * 8 + v;
    if (r < M) {
      float* o = out + (long)r * 64 + l16;
      o[0]  = c0[v] + bv0;
      o[16] = c1[v] + bv1;
      o[32] = c2[v] + bv2;
      o[48] = c3[v] + bv3;
    }
  }
}

// --------------------------- propagate: scatter-add ------------------------
// 16 lanes per edge, float4 gather (L2-resident src) + 4 fp32 atomics.

__global__ void __launch_bounds__(256)
scatter_kernel(const float* __restrict__ src, float* __restrict__ dst,
               const float* __restrict__ norm, const int* __restrict__ row,
               const int* __restrict__ col) {
  long t = (long)blockIdx.x * 256 + threadIdx.x;
  long e = t >> 4;
  if (e >= N_EDGES) return;
  int f0 = ((int)t & 15) << 2;
  float wv = norm[e];
  if (wv == 0.0f) return;  // ReLU'd edge weights: skip dead edges
  long r = row[e], c = col[e];
  float4 s = *reinterpret_cast<const float4*>(src + r * 64 + f0);
  float* d = dst + c * 64 + f0;
  atomicAdd(d + 0, wv * s.x);
  atomicAdd(d + 1, wv * s.y);
  atomicAdd(d + 2, wv * s.z);
  atomicAdd(d + 3, wv * s.w);
}

// --------------------------- launcher --------------------------------------

extern "C" void kernel_launch(void* const* d_in, const int* in_sizes, int n_in,
                              void* d_out, int out_size, void* d_ws,
                              size_t ws_size, hipStream_t stream) {
  const float* x      = (const float*)d_in[0];
  const int*   ei     = (const int*)d_in[1];
  const float* ea     = (const float*)d_in[2];
  const float* mw1    = (const float*)d_in[3];
  const float* mb1    = (const float*)d_in[4];
  const float* mw2    = (const float*)d_in[5];
  const float* mb2    = (const float*)d_in[6];
  const float* init_w = (const float*)d_in[7];
  const float* arma_w = (const float*)d_in[8];
  const float* rw0    = (const float*)d_in[9];
  const float* rw1    = (const float*)d_in[10];
  const float* b0     = (const float*)d_in[11];
  const float* b1     = (const float*)d_in[12];
  const float* lw     = (const float*)d_in[13];
  const float* lb     = (const float*)d_in[14];
  float* out = (float*)d_out;
  const int* rowi = ei;            // edge_index[0] : source
  const int* coli = ei + N_EDGES;  // edge_index[1] : target

  // workspace: ew/norm | deg/dinv | Bp (prop input) | Ba0 | Ba1 (~84MB)
  char* ws = (char*)d_ws;
  auto al = [](size_t v) { return (v + 255) & ~(size_t)255; };
  size_t o = 0;
  float* ew  = (float*)(ws + o); o = al(o + sizeof(float) * N_EDGES);
  float* deg = (float*)(ws + o); o = al(o + sizeof(float) * N_NODES);
  float* Bp  = (float*)(ws + o); o = al(o + sizeof(float) * (size_t)N_NODES * 64);
  float* Ba0 = (float*)(ws + o); o = al(o + sizeof(float) * (size_t)N_NODES * 64);
  float* Ba1 = (float*)(ws + o); o = al(o + sizeof(float) * (size_t)N_NODES * 64);
  (void)o; (void)ws_size; (void)in_sizes; (void)n_in; (void)out_size;

  const int nb_nodes  = (N_NODES + 255) / 256;
  const int nb_edges  = (N_EDGES + 255) / 256;
  const int nb_etiles = (N_EDGES / 16) / 8;              // 12500, exact
  const int gtiles    = (N_NODES + 15) / 16;             // 6250
  const int nb_gemm   = (gtiles + 7) / 8;                // 8 waves/block
  const int nb_scat   = (int)(((long)N_EDGES * 16 + 255) / 256);  // 100000

  // gcn_norm (no self loops): ew = edgeMLP(attr); deg = seg_sum(ew, col)
  zero_kernel<<<nb_nodes, 256, 0, stream>>>(deg, N_NODES);
  edge_mlp_kernel<<<nb_etiles, 256, 0, stream>>>(ea, mw1, mb1, mw2, mb2, coli,
                                                 ew, deg);
  dinv_kernel<<<nb_nodes, 256, 0, stream>>>(deg, N_NODES);
  norm_kernel<<<nb_edges, 256, 0, stream>>>(ew, deg, rowi, coli);

  // t=0: Bp = x@init_w ; Ba0 = x@root_w0 + b0 ; Ba1 = x@root_w1 + b1
  gemm_node_kernel<0><<<nb_gemm, 256, 0, stream>>>(x, init_w, nullptr, Bp,
                                                   N_NODES, 128);
  gemm_node_kernel<0><<<nb_gemm, 256, 0, stream>>>(x, rw0, b0, Ba0,
                                                   N_NODES, 128);
  gemm_node_kernel<0><<<nb_gemm, 256, 0, stream>>>(x, rw1, b1, Ba1,
                                                   N_NODES, 128);
  // Ba0 += scatter(norm * Bp[row])  -> h = relu(Ba0) fused into next GEMM
  scatter_kernel<<<nb_scat, 256, 0, stream>>>(Bp, Ba0, ew, rowi, coli);
  // t=1: Bp = relu(Ba0) @ arma_w
  gemm_node_kernel<1><<<nb_gemm, 256, 0, stream>>>(Ba0, arma_w, nullptr, Bp,
                                                   N_NODES, 64);
  scatter_kernel<<<nb_scat, 256, 0, stream>>>(Bp, Ba1, ew, rowi, coli);
  // head: out = relu(Ba1) @ lin_w + lin_b
  gemm_node_kernel<1><<<nb_gemm, 256, 0, stream>>>(Ba1, lw, lb, out,
                                                   N_NODES, 64);
}